// QAgentlayer_57664230916486
// MI455X (gfx1250) — compile-verified
//
#include <hip/hip_runtime.h>

// MI455X / gfx1250, wave32. 4-qubit VQC batch simulator.
// psi storage (per wave, 16 batch elements): lane n holds amplitudes m=0..7 of
// batch element n in regs ar/ai[0..7]; lane n+16 holds amplitudes m=8..15.
// This is exactly the V_WMMA_F32_16X16X4_F32 C/D (and, half-swapped, B) layout,
// so the per-layer fixed RY*RZ*CZ unitary is applied with WMMA complex GEMMs.
//
// Cross-half exchange (lane i <-> lane i^16) is done with v_permlanex16_b32
// (pure VALU) instead of ds_bpermute, eliminating LDS-pipe traffic and
// s_wait_dscnt stalls between the WMMA chains.

typedef __attribute__((ext_vector_type(2))) float v2f;
typedef __attribute__((ext_vector_type(8))) float v8f;

#define NL 5

static __device__ __forceinline__ float xor16(float v) {
    // v_permlanex16_b32 with identity in-half selects: lane i (lo half) gets
    // the value from lane i of the hi half and vice versa == shfl_xor(v, 16).
    const int i = __float_as_int(v);
    const int r = __builtin_amdgcn_permlanex16(
        i, i, 0x76543210, (int)0xfedcba98, false, false);
    return __int_as_float(r);
}

static __device__ __forceinline__ v8f wmma4(v2f a, v2f b, v8f c) {
    // D = A(16x4) * B(4x16) + C ; f32, 8-arg form:
    // (neg_a, A, neg_b, B, c_mod, C, reuse_a, reuse_b)
    return __builtin_amdgcn_wmma_f32_16x16x4_f32(false, a, false, b, (short)0, c,
                                                 false, false);
}

__global__ __launch_bounds__(256)
void qsim16_kernel(const float4* __restrict__ x4,
                   const float*  __restrict__ wscale,   // (5,4)
                   const float*  __restrict__ wvar,     // (5,8)
                   float2*       __restrict__ out)      // (B,2)
{
    const int  lane = threadIdx.x & 31;
    const int  wv   = threadIdx.x >> 5;
    const int  j    = lane & 15;          // A-row (out amp) == batch column in tile
    const bool hi   = lane >= 16;
    const int  tile = blockIdx.x * 8 + wv;
    const int  b    = tile * 16 + j;

    const float4 xx = x4[b];
    const float xv[4] = { xx.x, xx.y, xx.z, xx.w };

    // ---- state init: |0000> ----
    float ar[8], ai[8];
    #pragma unroll
    for (int r = 0; r < 8; ++r) { ar[r] = 0.f; ai[r] = 0.f; }
    ar[0] = hi ? 0.f : 1.f;

    const int kb = hi ? 2 : 0;                       // A/B intra-slice K base for this half
    const int j0 = (j >> 3) & 1, j1 = (j >> 2) & 1, j2 = (j >> 1) & 1, j3 = j & 1;
    const int   jbit[4] = { j0, j1, j2, j3 };
    const float sgn[4]  = { (float)(2*j0-1), (float)(2*j1-1),
                            (float)(2*j2-1), (float)(2*j3-1) };
    const float czs = ((j0 & j1) ^ (j1 & j2) ^ (j2 & j3) ^ (j3 & j0)) ? -1.f : 1.f;

    #pragma unroll
    for (int l = 0; l < NL; ++l) {
        // ================= RX(w_l,q * x_b,q) on each wire =================
        // wire 0 -> amp bit 3 (stride 8): crosses lane halves; uniform update:
        // new_r = c*v_r + s*(partner_i), new_i = c*v_i - s*(partner_r)
        {
            float s, c;
            __sincosf(0.5f * wscale[l*4 + 0] * xv[0], &s, &c);
            #pragma unroll
            for (int r = 0; r < 8; ++r) {
                const float pr = xor16(ar[r]);
                const float pi = xor16(ai[r]);
                const float nr = c * ar[r] + s * pi;
                const float ni = c * ai[r] - s * pr;
                ar[r] = nr; ai[r] = ni;
            }
        }
        // wires 1..3 (strides 4,2,1): pairs live inside one lane's 8 regs
        #pragma unroll
        for (int q = 1; q < 4; ++q) {
            const int st = 8 >> q;
            float s, c;
            __sincosf(0.5f * wscale[l*4 + q] * xv[q], &s, &c);
            #pragma unroll
            for (int r = 0; r < 8; ++r) {
                if (!(r & st)) {
                    const int p = r + st;
                    const float a_r = ar[r], a_i = ai[r];
                    const float b_r = ar[p], b_i = ai[p];
                    ar[r] = c * a_r + s * b_i;  ai[r] = c * a_i - s * b_r;
                    ar[p] = c * b_r + s * a_i;  ai[p] = c * b_i - s * a_r;
                }
            }
        }

        // ============ per-layer fixed unitary U = Dcz * Drz * (x)RY ============
        // Each lane builds its A-matrix row j at its 8 K columns.
        float cy[4], sy[4];
        #pragma unroll
        for (int q = 0; q < 4; ++q)
            __sincosf(0.5f * wvar[l*8 + q], &sy[q], &cy[q]);
        float phi = 0.f;
        #pragma unroll
        for (int q = 0; q < 4; ++q)
            phi += 0.5f * wvar[l*8 + 4 + q] * sgn[q];
        float cph, sph;
        __sincosf(phi, &sph, &cph);
        cph *= czs; sph *= czs;                 // fold CZ sign into the row phase

        v2f Ar[4], Ai[4], nAi[4];
        #pragma unroll
        for (int s = 0; s < 4; ++s) {
            float ur[2], ui[2];
            #pragma unroll
            for (int t = 0; t < 2; ++t) {
                const int k = 4*s + kb + t;
                const int kq[4] = { (k>>3)&1, (k>>2)&1, (k>>1)&1, k&1 };
                float R = 1.f;
                #pragma unroll
                for (int q = 0; q < 4; ++q) {
                    const float f = (jbit[q] == kq[q]) ? cy[q]
                                   : ((jbit[q] == 0) ? -sy[q] : sy[q]);
                    R *= f;
                }
                ur[t] = R * cph;
                ui[t] = R * sph;
            }
            v2f tr;  tr.x  =  ur[0]; tr.y  =  ur[1];
            v2f ti;  ti.x  =  ui[0]; ti.y  =  ui[1];
            v2f tni; tni.x = -ui[0]; tni.y = -ui[1];
            Ar[s] = tr; Ai[s] = ti; nAi[s] = tni;
        }

        // ---- build B slices (K=4s..4s+3 x N=16) from the half-split state ----
        v2f Br[4], Bi[4];
        #pragma unroll
        for (int s = 0; s < 4; ++s) {
            v2f br, bi;
            if (s < 2) {
                // K rows 0..7 live in lo lanes; hi lanes fetch via xor-16
                const float t0r = xor16(ar[4*s + 2]), t0i = xor16(ai[4*s + 2]);
                const float t1r = xor16(ar[4*s + 3]), t1i = xor16(ai[4*s + 3]);
                br.x = hi ? t0r : ar[4*s];     br.y = hi ? t1r : ar[4*s + 1];
                bi.x = hi ? t0i : ai[4*s];     bi.y = hi ? t1i : ai[4*s + 1];
            } else {
                // K rows 8..15 live in hi lanes; lo lanes fetch via xor-16
                const int rb = 4*s - 8;
                const float t0r = xor16(ar[rb]),     t0i = xor16(ai[rb]);
                const float t1r = xor16(ar[rb + 1]), t1i = xor16(ai[rb + 1]);
                br.x = hi ? ar[rb + 2] : t0r;  br.y = hi ? ar[rb + 3] : t1r;
                bi.x = hi ? ai[rb + 2] : t0i;  bi.y = hi ? ai[rb + 3] : t1i;
            }
            Br[s] = br; Bi[s] = bi;
        }

        // ---- complex GEMM on the matrix units: 16 x v_wmma_f32_16x16x4_f32 ----
        v8f cr = {0.f,0.f,0.f,0.f,0.f,0.f,0.f,0.f};
        v8f ci = {0.f,0.f,0.f,0.f,0.f,0.f,0.f,0.f};
        #pragma unroll
        for (int s = 0; s < 4; ++s) cr = wmma4(Ar[s],  Br[s], cr);   //  Ur*psi_r
        #pragma unroll
        for (int s = 0; s < 4; ++s) cr = wmma4(nAi[s], Bi[s], cr);   // -Ui*psi_i
        #pragma unroll
        for (int s = 0; s < 4; ++s) ci = wmma4(Ai[s],  Br[s], ci);   //  Ui*psi_r
        #pragma unroll
        for (int s = 0; s < 4; ++s) ci = wmma4(Ar[s],  Bi[s], ci);   //  Ur*psi_i

        // D layout == our storage layout: lane n amps 0..7, lane n+16 amps 8..15
        #pragma unroll
        for (int r = 0; r < 8; ++r) { ar[r] = cr[r]; ai[r] = ci[r]; }
    }

    // ================= observables <Z0Z1>, <Z2Z3> =================
    float left = 0.f, right = 0.f;
    #pragma unroll
    for (int r = 0; r < 8; ++r) {
        const int   m = r + (hi ? 8 : 0);
        const float p = ar[r]*ar[r] + ai[r]*ai[r];
        left  += (((m >> 3) ^ (m >> 2)) & 1) ? -p : p;   // (-1)^(w0+w1)
        right += (((m >> 1) ^  m      ) & 1) ? -p : p;   // (-1)^(w2+w3)
    }
    left  += xor16(left);
    right += xor16(right);
    if (!hi) {
        float2 o; o.x = left; o.y = right;
        out[b] = o;
    }
}

extern "C" void kernel_launch(void* const* d_in, const int* in_sizes, int n_in,
                              void* d_out, int out_size, void* d_ws, size_t ws_size,
                              hipStream_t stream) {
    (void)n_in; (void)out_size; (void)d_ws; (void)ws_size;
    const float* x      = (const float*)d_in[0];   // (B,4) f32
    const float* wscale = (const float*)d_in[1];   // (5,4) f32
    const float* wvar   = (const float*)d_in[2];   // (5,8) f32
    const int B = in_sizes[0] / 4;                 // 131072
    const int blocks = B / (16 * 8);               // 8 waves per block, 16 batch/wave
    qsim16_kernel<<<blocks, 256, 0, stream>>>(
        (const float4*)x, wscale, wvar, (float2*)d_out);
}